// SKA3D_22256520528387
// MI455X (gfx1250) — compile-verified
//
#include <hip/hip_runtime.h>

// SKA3D (27-tap spatially-varying grouped conv, circular pad) for MI455X gfx1250.
//
// Geometry (fixed): B=2, C=64, G=8, Cg=8, D=H=W=64.
// Bandwidth-bound: ~720 MB compulsory traffic -> ~31 us floor @ 23.3 TB/s.
// Math on the matrix pipe via banded-B reformulation along W:
//   out[c',p] = sum_{(di,hi)} sum_j X[c',j] * Bband[j,p],  j = p + wi + 1
// using V_WMMA_F32_16X16X4_F32 (exact f32).  A-matrix rows M=8..15 duplicate
// rows 0..7 (clamped channel index) so no masking is needed; the duplicated
// accumulator rows are simply never stored.

typedef __attribute__((ext_vector_type(2))) float v2f;
typedef __attribute__((ext_vector_type(8))) float v8f;

#define WAVES_PER_BLOCK 8
#define NUM_BLOCKS      2048
#define TILES_PER_WAVE  16
// total tiles = B*G*D*H*(W/16) = 2*8*64*64*4 = 262144 = 2048*8*16 (exact)

// per-wave LDS slice: x window [r=9][c=8][j=20(18+2 pad)] + w tile [k=27][p=16]
#define XS_FLOATS (9 * 8 * 20)   // 1440
#define WS_FLOATS (27 * 16)      // 432
#define SLICE     (XS_FLOATS + WS_FLOATS)

__global__ __launch_bounds__(256) void ska3d_wmma_kernel(
    const float* __restrict__ x,   // [2][64][64][64][64]
    const float* __restrict__ w,   // [2][8][27][64][64][64]
    float* __restrict__ out)       // [2][64][64][64][64]
{
    __shared__ float lds[WAVES_PER_BLOCK * SLICE];

    const int lane = threadIdx.x & 31;
    const int wv   = threadIdx.x >> 5;
    float* xsw = lds + wv * SLICE;
    float* wsw = xsw + XS_FLOATS;
    const unsigned ldsXsBase = (unsigned)(wv * SLICE) * 4u;  // byte offset of xsw in LDS

    const int half = lane >> 4;   // K-pair select {0,1} vs {2,3} (ISA wave32 layout)
    const int l15  = lane & 15;   // M (A/C) and N (B/C) index for this lane

    const int waveGlobal = blockIdx.x * WAVES_PER_BLOCK + wv;

    for (int it = 0; it < TILES_PER_WAVE; ++it) {
        const int tile = waveGlobal * TILES_PER_WAVE + it;  // consecutive -> x reuse
        const int w0 = (tile & 3) << 4;
        const int h  = (tile >> 2)  & 63;
        const int d  = (tile >> 8)  & 63;
        const int g  = (tile >> 14) & 7;
        const int b  = (tile >> 17) & 1;

        const float* xb = x + (size_t)(b * 64 + g * 8) * 262144;
        const float* wb = w + (size_t)(b * 8 + g) * 27 * 262144
                            + (size_t)(d * 4096 + h * 64 + w0);

        // ---- async-copy x window (9 rows x 8 ch x 18 w) straight into LDS
        for (int t = lane; t < 1296; t += 32) {
            const int j = t % 18;
            const int c = (t / 18) & 7;
            const int r = t / 144;                    // r = (di+1)*3 + (hi+1)
            const int dd = (d + (r / 3) + 63) & 63;   // (d + di) mod 64
            const int hh = (h + (r % 3) + 63) & 63;   // (h + hi) mod 64
            const int ww = (w0 + j + 63) & 63;        // (w0 - 1 + j) mod 64
            const unsigned long long gaddr = (unsigned long long)(const void*)
                (xb + (size_t)c * 262144 + (size_t)(dd * 4096 + hh * 64 + ww));
            const unsigned laddr = ldsXsBase + (unsigned)(r * 160 + c * 20 + j) * 4u;
            asm volatile("global_load_async_to_lds_b32 %0, %1, off"
                         :: "v"(laddr), "v"(gaddr) : "memory");
        }
        // zero the j=18,19 K-padding (plain DS stores; disjoint from async targets)
        for (int t = lane; t < 144; t += 32) {
            const int r = t >> 4;
            const int c = (t >> 1) & 7;
            xsw[r * 160 + c * 20 + 18 + (t & 1)] = 0.0f;
        }
        // ---- stage w tile: 27 taps x 16 positions, stream-once -> non-temporal
        for (int t = lane; t < 432; t += 32) {
            const int p = t & 15;
            const int k = t >> 4;
            wsw[t] = __builtin_nontemporal_load(wb + (size_t)k * 262144 + p);
        }
        // ---- prefetch next tile's weight cachelines (global_prefetch_b8)
        if (it + 1 < TILES_PER_WAVE) {
            const int tn  = tile + 1;
            const int w0n = (tn & 3) << 4;
            const int hn  = (tn >> 2)  & 63;
            const int dn  = (tn >> 8)  & 63;
            const int gn  = (tn >> 14) & 7;
            const int bn  = (tn >> 17) & 1;
            const float* wbn = w + (size_t)(bn * 8 + gn) * 27 * 262144
                                 + (size_t)(dn * 4096 + hn * 64 + w0n);
            for (int t = lane; t < 27; t += 32)
                __builtin_prefetch(wbn + (size_t)t * 262144, 0, 0);
        }
        asm volatile("s_wait_asynccnt 0x0" ::: "memory");
        __syncthreads();

        // ---- 45x V_WMMA_F32_16X16X4_F32 (5 K-chunks x 9 (di,hi) rows)
        v8f acc = {};
        const float* XmBase = xsw + (l15 & 7) * 20 + 2 * half;  // clamped channel row

        #pragma unroll
        for (int r = 0; r < 9; ++r) {
            const float* Xm = XmBase + r * 160;
            // unconditional contiguous A fragments: one ds_load_b64 each
            v2f afr[5];
            #pragma unroll
            for (int q = 0; q < 5; ++q)
                afr[q] = *(const v2f*)(Xm + 4 * q);
            // the three W-taps of this (di,hi) at output column p = l15
            const float wm1 = wsw[(r * 3 + 0) * 16 + l15];  // wi = -1
            const float w0v = wsw[(r * 3 + 1) * 16 + l15];  // wi =  0
            const float wp1 = wsw[(r * 3 + 2) * 16 + l15];  // wi = +1
            // banded B fragments: nonzero iff j - p == wi + 1 in {0,1,2}
            v2f bfr[5];
            #pragma unroll
            for (int q = 0; q < 5; ++q) {
                const int r0 = 4 * q + 2 * half - l15;
                bfr[q].x = (r0 == 0) ? wm1 : (r0 == 1) ? w0v : (r0 == 2) ? wp1 : 0.0f;
                const int r1 = r0 + 1;
                bfr[q].y = (r1 == 0) ? wm1 : (r1 == 1) ? w0v : (r1 == 2) ? wp1 : 0.0f;
            }
            #pragma unroll
            for (int q = 0; q < 5; ++q)
                acc = __builtin_amdgcn_wmma_f32_16x16x4_f32(
                        false, afr[q], false, bfr[q], (short)0, acc, false, false);
        }

        // ---- store: C/D layout -> lanes 0..15 hold M=r (=channel r) at N=l15
        if (half == 0) {
            float* ob = out + (size_t)(b * 64 + g * 8) * 262144
                            + (size_t)(d * 4096 + h * 64 + w0 + l15);
            #pragma unroll
            for (int r = 0; r < 8; ++r)
                __builtin_nontemporal_store(acc[r], ob + (size_t)r * 262144);
        }
    }
}

extern "C" void kernel_launch(void* const* d_in, const int* in_sizes, int n_in,
                              void* d_out, int out_size, void* d_ws, size_t ws_size,
                              hipStream_t stream) {
    (void)in_sizes; (void)n_in; (void)out_size; (void)d_ws; (void)ws_size;
    const float* x = (const float*)d_in[0];
    const float* w = (const float*)d_in[1];
    float* out = (float*)d_out;
    ska3d_wmma_kernel<<<dim3(NUM_BLOCKS), dim3(WAVES_PER_BLOCK * 32), 0, stream>>>(x, w, out);
}